// LocallyConnected1d_37830071943927
// MI455X (gfx1250) — compile-verified
//
#include <hip/hip_runtime.h>
#include <hip/hip_bf16.h>

// Locally-connected 1D layer on gfx1250 (MI455X), FP32 WMMA path.
// Per output position l: GEMM M=64(batch) x K=576(cin*ktap) x N=64(cout),
// weights unique per l. Memory-bound (~92MB @ 23.3TB/s); f32 16x16x4 WMMA
// preserves reference precision and is far from the compute ceiling.

typedef float v2f __attribute__((ext_vector_type(2)));
typedef float v8f __attribute__((ext_vector_type(8)));

#define BATCH 64
#define CIN   64
#define COUT  64
#define SEQL  512
#define KS    9
#define PAD   4
#define KTOT  (CIN * KS)     // 576
#define KC    144            // K-chunk: 16 input channels * 9 taps
#define NCHUNK (KTOT / KC)   // 4
#define A_STRIDE 148         // padded row stride (floats): even (b64 align), bank-skewed
#define B_STRIDE 66          // padded row stride (floats): bank-skewed

__global__ __launch_bounds__(512)
void lc1d_wmma_f32(const float* __restrict__ x,
                   const float* __restrict__ w,
                   const float* __restrict__ bias,
                   float* __restrict__ out)
{
    __shared__ __align__(16) float lds_a[BATCH * A_STRIDE]; // 37.9 KB: patches[m][kk]
    __shared__ __align__(16) float lds_b[KC * B_STRIDE];    // 38.0 KB: weights[kk][o]

    const int l    = blockIdx.x;        // one workgroup per output position
    const int tid  = threadIdx.x;       // 512 threads = 16 wave32
    const int wave = tid >> 5;
    const int lane = tid & 31;
    const int half = lane >> 4;         // 0: lanes 0-15, 1: lanes 16-31
    const int lr   = lane & 15;

    const int m0 = (wave >> 2) * 16;    // batch-tile origin
    const int n0 = (wave & 3) * 16;     // outch-tile origin

    v8f acc = {};                       // 16x16 f32 accumulator (8 VGPRs)

    for (int c = 0; c < NCHUNK; ++c) {
        // ---- stage A chunk: patches[m][r], r = (i_local*9 + k), i = 16c + i_local
        for (int idx = tid; idx < BATCH * KC; idx += 512) {
            const int m  = idx / KC;
            const int r  = idx - m * KC;
            const int il = r / 9;
            const int k  = r - il * 9;
            const int t  = l + k - PAD;
            float v = 0.0f;
            if (t >= 0 && t < SEQL)
                v = x[(m * CIN + (c * 16 + il)) * SEQL + t];
            lds_a[m * A_STRIDE + r] = v;
        }
        // ---- stage B chunk: W[kk][o] = weight[i][o][k][l]
        for (int idx = tid; idx < KC * COUT; idx += 512) {
            const int kk = idx >> 6;
            const int o  = idx & 63;
            const int il = kk / 9;
            const int k  = kk - il * 9;
            const float* wp = &w[(((c * 16 + il) * COUT + o) * KS + k) * SEQL + l];
            lds_b[kk * B_STRIDE + o] = *wp;
            if (c + 1 < NCHUNK)  // prefetch next chunk's weights (global_prefetch)
                __builtin_prefetch(wp + 16 * COUT * KS * SEQL, 0, 0);
        }
        __syncthreads();

        // ---- 36 WMMA k-steps from LDS (EXEC all-ones here)
        #pragma unroll 4
        for (int kk0 = 0; kk0 < KC; kk0 += 4) {
            const int ka = kk0 + half * 2;
            // A 16x4 fragment: lane lr holds A[m0+lr][ka], A[m0+lr][ka+1]  (8B ds load)
            v2f a = *reinterpret_cast<const v2f*>(&lds_a[(m0 + lr) * A_STRIDE + ka]);
            // B 4x16 fragment: lane lr holds B[ka][n0+lr], B[ka+1][n0+lr]
            v2f b;
            b.x = lds_b[(ka    ) * B_STRIDE + n0 + lr];
            b.y = lds_b[(ka + 1) * B_STRIDE + n0 + lr];
            acc = __builtin_amdgcn_wmma_f32_16x16x4_f32(
                      /*neg_a=*/false, a, /*neg_b=*/false, b,
                      /*c_mod=*/(short)0, acc, /*reuse_a=*/false, /*reuse_b=*/false);
        }
        __syncthreads();
    }

    // ---- epilogue: C/D layout: VGPR r -> M = m0 + r + half*8, N = n0 + lr
    const int o  = n0 + lr;
    const float bv = bias[o * SEQL + l];
    #pragma unroll
    for (int r = 0; r < 8; ++r) {
        const int m = m0 + r + half * 8;
        out[(m * COUT + o) * SEQL + l] = acc[r] + bv;
    }
}

extern "C" void kernel_launch(void* const* d_in, const int* in_sizes, int n_in,
                              void* d_out, int out_size, void* d_ws, size_t ws_size,
                              hipStream_t stream) {
    (void)in_sizes; (void)n_in; (void)out_size; (void)d_ws; (void)ws_size;
    const float* x    = (const float*)d_in[0];
    const float* wgt  = (const float*)d_in[1];
    const float* bias = (const float*)d_in[2];
    float* out        = (float*)d_out;
    lc1d_wmma_f32<<<SEQL, 512, 0, stream>>>(x, wgt, bias, out);
}